// TwoTaskNetWithLoss_33861522162044
// MI455X (gfx1250) — compile-verified
//
#include <hip/hip_runtime.h>
#include <hip/hip_bf16.h>

#define IMG  256
#define THRE 0.5f
#define DIS  4

// ---------------- CDNA5 async global->LDS path (guarded) ----------------
#if defined(__has_builtin)
#  if __has_builtin(__builtin_amdgcn_global_load_async_to_lds_b32)
#    define USE_ASYNC_LDS 1
#  endif
#endif

#if defined(USE_ASYNC_LDS) && __has_builtin(__builtin_amdgcn_s_wait_asynccnt)
#  define WAIT_ASYNC() __builtin_amdgcn_s_wait_asynccnt(0)
#elif defined(USE_ASYNC_LDS)
#  define WAIT_ASYNC() asm volatile("s_wait_asynccnt 0x0" ::: "memory")
#else
#  define WAIT_ASYNC() ((void)0)
#endif

__device__ __forceinline__ void async_copy_b32(const float* src, float* dst_lds) {
#ifdef USE_ASYNC_LDS
  // (global src, lds dst, imm offset, imm cpol) -> global_load_async_to_lds_b32
  __builtin_amdgcn_global_load_async_to_lds_b32(
      (__attribute__((address_space(1))) int*)(const_cast<float*>(src)),
      (__attribute__((address_space(3))) int*)(dst_lds),
      0, 0);
#else
  *dst_lds = *src;   // fallback: normal load + ds store
#endif
}

// ---------------- crops kernel: per-cell min/max normalize * mask ----------------
// Block = 256 threads (one per output column), grid = (G cell-rows, 128 batch-slots).
// bb < 64  : plain img cells.  bb >= 64 : cells shifted up-left by DIS (clamped at 0).
template <int G>
__global__ void crops_kernel(const float* __restrict__ img,
                             const float* __restrict__ heat,
                             float* __restrict__ out) {
  constexpr int S = IMG / G;                 // cell size: 8, 16, or 32
  __shared__ float band[S * IMG];            // one cell-row band, max 32 KB

  const int t  = threadIdx.x;                // output column 0..255
  const int r  = blockIdx.x;                 // cell row 0..G-1
  const int bb = blockIdx.y;                 // 0..127
  const int b  = bb & 63;                    // underlying img batch
  const bool shifted = bb >= 64;

  const int c  = t / S;                      // cell column
  const int jj = t % S;                      // offset within cell
  int base_r = r * S;
  int base_c = c * S;
  if (shifted) {
    base_r = (base_r - DIS > 0) ? base_r - DIS : 0;
    base_c = (base_c - DIS > 0) ? base_c - DIS : 0;
  }

  const float* src = img + (size_t)b * (IMG * IMG) + (size_t)base_r * IMG + (base_c + jj);

  // Stage this thread's source column (S rows) into LDS via async copies.
#pragma unroll
  for (int i = 0; i < S; ++i)
    async_copy_b32(src + i * IMG, &band[i * IMG + t]);
  WAIT_ASYNC();
  // No barrier needed: each thread only reads back its own column, and each
  // cell's S lanes live inside one wave32 (S in {8,16,32} aligned groups).

  // Pass 1: per-column min/max, then reduce across the S lanes of the cell.
  float mn = band[t], mx = mn;
#pragma unroll
  for (int i = 1; i < S; ++i) {
    const float v = band[i * IMG + t];
    mn = fminf(mn, v);
    mx = fmaxf(mx, v);
  }
#pragma unroll
  for (int off = S >> 1; off > 0; off >>= 1) {
    mn = fminf(mn, __shfl_xor(mn, off, 32));
    mx = fmaxf(mx, __shfl_xor(mx, off, 32));
  }

  const float m   = (heat[(size_t)bb * (G * G) + r * G + c] > THRE) ? 1.0f : 0.0f;
  const float inv = m / (mx - mn + 1e-5f);   // m==0 -> output exactly 0

  // Pass 2: normalize from LDS, streaming NT stores (output never re-read).
  float* dst = out + (size_t)bb * (IMG * IMG) + (size_t)(r * S) * IMG + t;
#pragma unroll
  for (int i = 0; i < S; ++i)
    __builtin_nontemporal_store((band[i * IMG + t] - mn) * inv, dst + i * IMG);
}

// ---------------- label kernel: scatter inverted into a gather ----------------
// inverse of Y[p] = max(p//S*S - DIS, 0) + p%S :
//   target t receives source t     iff t < S
//   target t receives source t+DIS iff (t >= S-DIS) && (t+DIS < IMG)
template <int G>
__device__ __forceinline__ void accum_scale(const float* __restrict__ heat,
                                            const float* __restrict__ seg,
                                            int b, int ti, int tj,
                                            float& num, float& den) {
  constexpr int S = IMG / G;
  // direct part (ca * seg[:B])
  if (heat[(size_t)b * (G * G) + (ti / S) * G + (tj / S)] > THRE) {
    num += seg[(size_t)b * (IMG * IMG) + ti * IMG + tj];
    den += 1.0f;
  }
  // gathered shifted part (cd * seg[B:])
  int pis[2]; int npi = 0;
  if (ti < S)                              pis[npi++] = ti;
  if (ti >= S - DIS && ti + DIS < IMG)     pis[npi++] = ti + DIS;
  int pjs[2]; int npj = 0;
  if (tj < S)                              pjs[npj++] = tj;
  if (tj >= S - DIS && tj + DIS < IMG)     pjs[npj++] = tj + DIS;

  const float* heatD = heat + (size_t)(b + 64) * (G * G);
  const float* segD  = seg  + (size_t)(b + 64) * (IMG * IMG);
  for (int a = 0; a < npi; ++a)
    for (int q = 0; q < npj; ++q) {
      const int pi = pis[a], pj = pjs[q];
      if (heatD[(pi / S) * G + (pj / S)] > THRE) {
        num += segD[pi * IMG + pj];
        den += 1.0f;
      }
    }
}

__global__ void label_kernel(const float* __restrict__ h8,  const float* __restrict__ h16,
                             const float* __restrict__ h32,
                             const float* __restrict__ s8,  const float* __restrict__ s16,
                             const float* __restrict__ s32,
                             float* __restrict__ out) {
  const int tj = threadIdx.x;   // column
  const int ti = blockIdx.x;    // row
  const int b  = blockIdx.y;    // batch 0..63
  float num = 0.0f, den = 0.0f;
  accum_scale<32>(h32, s32, b, ti, tj, num, den);
  accum_scale<16>(h16, s16, b, ti, tj, num, den);
  accum_scale<8> (h8,  s8,  b, ti, tj, num, den);
  __builtin_nontemporal_store(num / (den + 1e-10f),
                              out + (size_t)b * (IMG * IMG) + ti * IMG + tj);
}

// ---------------- launch ----------------
extern "C" void kernel_launch(void* const* d_in, const int* in_sizes, int n_in,
                              void* d_out, int out_size, void* d_ws, size_t ws_size,
                              hipStream_t stream) {
  (void)in_sizes; (void)n_in; (void)out_size; (void)d_ws; (void)ws_size;
  const float* img = (const float*)d_in[0];
  const float* h8  = (const float*)d_in[1];
  const float* h16 = (const float*)d_in[2];
  const float* h32 = (const float*)d_in[3];
  const float* s8  = (const float*)d_in[4];
  const float* s16 = (const float*)d_in[5];
  const float* s32 = (const float*)d_in[6];

  float* out = (float*)d_out;
  const size_t IMG2 = (size_t)IMG * IMG;
  float* crops32 = out;                 // 128 * 256*256
  float* crops16 = out + 128 * IMG2;    // 128 * 256*256
  float* crops8  = out + 256 * IMG2;    // 128 * 256*256
  float* label   = out + 384 * IMG2;    //  64 * 256*256

  crops_kernel<32><<<dim3(32, 128), 256, 0, stream>>>(img, h32, crops32);
  crops_kernel<16><<<dim3(16, 128), 256, 0, stream>>>(img, h16, crops16);
  crops_kernel<8> <<<dim3(8, 128),  256, 0, stream>>>(img, h8,  crops8);
  label_kernel<<<dim3(256, 64), 256, 0, stream>>>(h8, h16, h32, s8, s16, s32, label);
}